// fix_text_img_32066225832156
// MI455X (gfx1250) — compile-verified
//
#include <hip/hip_runtime.h>
#include <hip/hip_bf16.h>
#include <stdint.h>

#define PAD_TOKEN_ID 0
#define IMAGE_TOKEN_INDEX 32000
#define IGNORE_INDEX (-100)

// problem constants (fixed by the reference)
#define BB 8
#define LL 4096
#define KK 576
#define DD 4096
#define ROW_BYTES (DD * 4)          // 16384 bytes per (b,l) row

// ---------------------------------------------------------------------------
// Kernel 1: per-row metadata (prefix scans) + the four small outputs.
// One block per batch row b; 256 threads; each thread owns 16 consecutive
// tokens. Two block-level scans: (1) image-token count -> rank base,
// (2) final attention mask -> position_ids cumsum.
// code[b*L+l]: >=0 -> image feature row (rank), -2 -> passthrough, -1 -> zero
// ---------------------------------------------------------------------------
__global__ __launch_bounds__(256)
void meta_kernel(const int* __restrict__ input_ids,
                 const int* __restrict__ attn,
                 const int* __restrict__ labels,
                 float* __restrict__ out_att,
                 float* __restrict__ out_lab,
                 float* __restrict__ out_pos,
                 float* __restrict__ out_imask,
                 int* __restrict__ code) {
    const int b = blockIdx.x;
    const int t = threadIdx.x;                 // 0..255
    const int base = b * LL + t * 16;          // 16 tokens per thread

    __shared__ int s_scan[256];

    // pass 1: load ids, count image tokens in my chunk
    int ids[16];
    int cnt = 0;
    #pragma unroll
    for (int i = 0; i < 16; ++i) {
        ids[i] = input_ids[base + i];
        cnt += (ids[i] == IMAGE_TOKEN_INDEX) ? 1 : 0;
    }
    s_scan[t] = cnt;
    __syncthreads();
    if (t == 0) {                              // serial exclusive scan of 256 partials
        int acc = 0;
        for (int i = 0; i < 256; ++i) { int v = s_scan[i]; s_scan[i] = acc; acc += v; }
    }
    __syncthreads();
    const int rank0 = s_scan[t];               // #image tokens before my chunk
    __syncthreads();

    // pass 2: per-element attention value, local sum for position_ids cumsum
    int attv[16];
    int attsum = 0;
    {
        int r = rank0;
        #pragma unroll
        for (int i = 0; i < 16; ++i) {
            const bool is_img = (ids[i] == IMAGE_TOKEN_INDEX);
            const bool wr = is_img && (r < KK);
            const int a = wr ? 1 : (is_img ? 0 : attn[base + i]);
            attv[i] = a;
            attsum += a;
            if (is_img) r++;
        }
    }
    s_scan[t] = attsum;
    __syncthreads();
    if (t == 0) {
        int acc = 0;
        for (int i = 0; i < 256; ++i) { int v = s_scan[i]; s_scan[i] = acc; acc += v; }
    }
    __syncthreads();
    int acum = s_scan[t];                      // att cumsum before my chunk

    // pass 3: emit small outputs + routing code
    int r = rank0;
    #pragma unroll
    for (int i = 0; i < 16; ++i) {
        const int gi = base + i;
        const bool is_img = (ids[i] == IMAGE_TOKEN_INDEX);
        const bool wr = is_img && (r < KK);
        const bool ex = is_img && !wr;
        acum += attv[i];
        int pos = acum - 1; if (pos < 0) pos = 0;
        out_att[gi]   = (float)attv[i];
        out_lab[gi]   = is_img ? (float)IGNORE_INDEX : (float)labels[gi];
        out_pos[gi]   = (float)pos;
        out_imask[gi] = wr ? 1.0f : 0.0f;      // final_input_ids==32000 <=> write
        code[gi]      = wr ? r : (ex ? -1 : -2);
        if (is_img) r++;
    }
}

// ---------------------------------------------------------------------------
// Kernel 2: 16KB row router, one block (64 threads = 2 waves) per (b,l) row.
// Copy rows use the CDNA5 async data mover:
//   16x global_load_async_to_lds_b128  ->  s_wait_asynccnt 0  ->
//   16x global_store_async_from_lds_b128
// Each wave32 keeps 8KB (16 x 512B) of async loads in flight; with 16KB LDS
// per block, a WGP fills all 320KB of LDS with in-flight data (20 blocks /
// 40 waves resident), maximizing latency hiding for the global->LDS->global
// pipeline. Waves own disjoint LDS slices, so only the wave-local ASYNCcnt
// wait is needed (no barriers).
// ---------------------------------------------------------------------------
__global__ __launch_bounds__(64)
void embed_kernel(const float* __restrict__ image_features,
                  const float* __restrict__ inputs_embeds,
                  const int* __restrict__ code,
                  float* __restrict__ out) {
    const int row = blockIdx.x;                // b*L + l
    const int b = row >> 12;                   // row / 4096
    const int c = code[row];
    float* dst = out + (size_t)row * DD;
    const int t = threadIdx.x;                 // 0..63

    __shared__ __align__(16) char lbuf[ROW_BYTES]; // one full row staged in LDS

    if (c == -1) {
        // zero-fill row: plain b128 stores (4096 floats / 64 threads)
        float4 z = make_float4(0.f, 0.f, 0.f, 0.f);
        float4* d4 = (float4*)dst;
        #pragma unroll
        for (int k = 0; k < 16; ++k) d4[t + k * 64] = z;
        return;
    }

    const float* src = (c == -2)
        ? (inputs_embeds + (size_t)row * DD)
        : (image_features + ((size_t)b * KK + (size_t)c) * DD);

    // LDS aperture: generic pointer's low 32 bits are the LDS byte offset
    const uint32_t lds_base = (uint32_t)(uintptr_t)(void*)lbuf;
    const uint32_t off0 = (uint32_t)t * 16u;   // 16 bytes per lane per op

    // burst 16 async b128 loads: this wave keeps 8KB outstanding
    #pragma unroll
    for (int k = 0; k < 16; ++k) {
        const uint32_t o = off0 + (uint32_t)k * 1024u;
        const uint64_t ga = (uint64_t)(uintptr_t)src + o;
        const uint32_t la = lds_base + o;
        asm volatile("global_load_async_to_lds_b128 %0, %1, off"
                     :: "v"(la), "v"(ga) : "memory");
    }
    // wave-local wait for the async loads to land in LDS
    asm volatile("s_wait_asynccnt 0" ::: "memory");

    // burst 16 async b128 stores LDS -> output row
    #pragma unroll
    for (int k = 0; k < 16; ++k) {
        const uint32_t o = off0 + (uint32_t)k * 1024u;
        const uint64_t ga = (uint64_t)(uintptr_t)dst + o;
        const uint32_t la = lds_base + o;
        asm volatile("global_store_async_from_lds_b128 %0, %1, off"
                     :: "v"(ga), "v"(la) : "memory");
    }
    asm volatile("s_wait_asynccnt 0" ::: "memory");
}

// ---------------------------------------------------------------------------
// launch
// ---------------------------------------------------------------------------
extern "C" void kernel_launch(void* const* d_in, const int* in_sizes, int n_in,
                              void* d_out, int out_size, void* d_ws, size_t ws_size,
                              hipStream_t stream) {
    const float* image_features = (const float*)d_in[0];   // (B,K,D) f32
    const float* inputs_embeds  = (const float*)d_in[1];   // (B,L,D) f32
    const int*   input_ids      = (const int*)d_in[2];     // (B,L)
    const int*   attn           = (const int*)d_in[3];     // (B,L)
    const int*   labels         = (const int*)d_in[4];     // (B,L)

    float* out = (float*)d_out;
    float* out_emb   = out;                                   // B*L*D
    float* out_att   = out + (size_t)BB * LL * DD;            // B*L
    float* out_lab   = out_att + (size_t)BB * LL;             // B*L
    float* out_pos   = out_lab + (size_t)BB * LL;             // B*L
    float* out_imask = out_pos + (size_t)BB * LL;             // B*L

    int* code = (int*)d_ws;                                   // B*L ints (128 KB)

    meta_kernel<<<BB, 256, 0, stream>>>(input_ids, attn, labels,
                                        out_att, out_lab, out_pos, out_imask,
                                        code);
    embed_kernel<<<BB * LL, 64, 0, stream>>>(image_features, inputs_embeds,
                                             code, out_emb);
}